// PytorchMeanshift_77103252898444
// MI455X (gfx1250) — compile-verified
//
#include <hip/hip_runtime.h>
#include <hip/hip_bf16.h>

#define NPTS 6144
#define FEA  32
#define NBW  4
#define NITER 4
#define BN_EPS 1e-5f

typedef float v2f __attribute__((ext_vector_type(2)));
typedef float v8f __attribute__((ext_vector_type(8)));

// ---------------- zero helper ----------------
__global__ void ms_zero_kernel(float* __restrict__ p, int n) {
    int i = blockIdx.x * blockDim.x + threadIdx.x;
    if (i < n) p[i] = 0.0f;
}

// ---------------- pack X -> float4(x,y,z,|x|^2) ----------------
__global__ void ms_pack_kernel(const float* __restrict__ X, float4* __restrict__ Xs) {
    int i = blockIdx.x * blockDim.x + threadIdx.x;
    if (i < NPTS) {
        float x = X[3 * i + 0], y = X[3 * i + 1], z = X[3 * i + 2];
        Xs[i] = make_float4(x, y, z, x * x + y * y + z * z);
    }
}

// ---------------- MLP stage 1: h = X_fea @ W1 + b1, accumulate batch sums ----------------
__global__ void ms_mlp1_kernel(const float* __restrict__ Xfea,
                               const float* __restrict__ W1,
                               const float* __restrict__ b1,
                               float* __restrict__ h,
                               float* __restrict__ sums /* [64]: sum[32], sumsq[32] */) {
    __shared__ float s_sum[FEA];
    __shared__ float s_sq[FEA];
    int t = threadIdx.x;
    if (t < FEA) { s_sum[t] = 0.0f; s_sq[t] = 0.0f; }
    __syncthreads();

    int row = blockIdx.x * blockDim.x + t;
    float x[FEA];
    float acc[FEA];
#pragma unroll
    for (int k = 0; k < FEA; k++) x[k] = Xfea[row * FEA + k];
#pragma unroll
    for (int j = 0; j < FEA; j++) acc[j] = b1[j];
    for (int k = 0; k < FEA; k++) {
        float xk = x[k];
#pragma unroll
        for (int j = 0; j < FEA; j++) acc[j] = fmaf(xk, W1[k * FEA + j], acc[j]);
    }
#pragma unroll
    for (int j = 0; j < FEA; j++) {
        h[row * FEA + j] = acc[j];
        atomicAdd(&s_sum[j], acc[j]);
        atomicAdd(&s_sq[j], acc[j] * acc[j]);
    }
    __syncthreads();
    if (t < FEA) {
        atomicAdd(&sums[t], s_sum[t]);
        atomicAdd(&sums[FEA + t], s_sq[t]);
    }
}

// ---------------- MLP stage 2: batchnorm -> relu -> @W2+b2 -> normalized softmax ----------------
__global__ void ms_mlp2_kernel(const float* __restrict__ h,
                               const float* __restrict__ sums,
                               const float* __restrict__ gamma,
                               const float* __restrict__ beta,
                               const float* __restrict__ W2,
                               const float* __restrict__ b2,
                               float* __restrict__ w) {
    int row = blockIdx.x * blockDim.x + threadIdx.x;
    float lg[NBW];
#pragma unroll
    for (int b = 0; b < NBW; b++) lg[b] = b2[b];
#pragma unroll
    for (int j = 0; j < FEA; j++) {
        float mu  = sums[j] * (1.0f / NPTS);
        float var = sums[FEA + j] * (1.0f / NPTS) - mu * mu;
        float hv  = (h[row * FEA + j] - mu) * rsqrtf(var + BN_EPS) * gamma[j] + beta[j];
        hv = fmaxf(hv, 0.0f);
#pragma unroll
        for (int b = 0; b < NBW; b++) lg[b] = fmaf(hv, W2[j * NBW + b], lg[b]);
    }
    float m  = fmaxf(fmaxf(lg[0], lg[1]), fmaxf(lg[2], lg[3]));
    float e0 = __expf(lg[0] - m), e1 = __expf(lg[1] - m);
    float e2 = __expf(lg[2] - m), e3 = __expf(lg[3] - m);
    float inv = 1.0f / (e0 + e1 + e2 + e3);   // folds softmax denom AND the final /sum(w)
    w[row * NBW + 0] = e0 * inv;
    w[row * NBW + 1] = e1 * inv;
    w[row * NBW + 2] = e2 * inv;
    w[row * NBW + 3] = e3 * inv;
}

// ---------------- meanshift step: one wave per (i-tile, bandwidth) ----------------
// Gram tile via V_WMMA_F32_16X16X4_F32 (K = D padded 3->4).
__global__ void __launch_bounds__(256)
ms_step_kernel(const float4* __restrict__ Xs,
               const float* __restrict__ w,
               float* __restrict__ Xout) {
    int lane = threadIdx.x & 31;
    int wid  = (blockIdx.x * blockDim.x + threadIdx.x) >> 5;
    int itile = wid >> 2;          // 0..383
    int band  = wid & 3;           // 0..3
    float bw2 = 0.25f * (float)(1 << (2 * band));   // 0.25, 1, 4, 16

    int  i0    = itile * 16;
    int  jn    = lane & 15;
    bool lo    = lane < 16;
    int  mbase = lo ? 0 : 8;

    // A matrix 16x4 (rows = i-tile points, K = xyz0): lanes 0-15 carry K=0,1; 16-31 carry K=2,3
    float4 pi = Xs[i0 + jn];
    v2f A;
    A.x = lo ? pi.x : pi.z;
    A.y = lo ? pi.y : 0.0f;

    float sqi[8];
#pragma unroll
    for (int r = 0; r < 8; r++) sqi[r] = Xs[i0 + mbase + r].w;

    float cnt[8], ax[8], ay[8], az[8];
#pragma unroll
    for (int r = 0; r < 8; r++) { cnt[r] = 0.0f; ax[r] = 0.0f; ay[r] = 0.0f; az[r] = 0.0f; }

    for (int j0 = 0; j0 < NPTS; j0 += 16) {
        float4 pj = Xs[j0 + jn];
        // B matrix 4x16 (K rows = xyz0, cols = j-tile points), same lane striping as A
        v2f B;
        B.x = lo ? pj.x : pj.z;
        B.y = lo ? pj.y : 0.0f;
        v8f c = {0.0f, 0.0f, 0.0f, 0.0f, 0.0f, 0.0f, 0.0f, 0.0f};
        c = __builtin_amdgcn_wmma_f32_16x16x4_f32(false, A, false, B,
                                                  (short)0, c, false, false);
        // lane holds gram[mbase+r][jn] in c[r]
        float sqj = pj.w;
#pragma unroll
        for (int r = 0; r < 8; r++) {
            float d2  = sqi[r] + sqj - 2.0f * c[r];
            float ind = (d2 <= bw2) ? 1.0f : 0.0f;
            cnt[r] += ind;
            ax[r] = fmaf(ind, pj.x, ax[r]);
            ay[r] = fmaf(ind, pj.y, ay[r]);
            az[r] = fmaf(ind, pj.z, az[r]);
        }
    }

    // reduce across the 16 lanes sharing each half (masks stay within the half)
#pragma unroll
    for (int r = 0; r < 8; r++) {
#pragma unroll
        for (int msk = 1; msk < 16; msk <<= 1) {
            cnt[r] += __shfl_xor(cnt[r], msk, 32);
            ax[r]  += __shfl_xor(ax[r],  msk, 32);
            ay[r]  += __shfl_xor(ay[r],  msk, 32);
            az[r]  += __shfl_xor(az[r],  msk, 32);
        }
    }

    if (jn == 0) {
#pragma unroll
        for (int r = 0; r < 8; r++) {
            int   row = i0 + mbase + r;
            float wb  = w[row * NBW + band];
            float inv = wb / cnt[r];
            atomicAdd(&Xout[row * 3 + 0], ax[r] * inv);
            atomicAdd(&Xout[row * 3 + 1], ay[r] * inv);
            atomicAdd(&Xout[row * 3 + 2], az[r] * inv);
        }
    }
}

// ---------------- host ----------------
extern "C" void kernel_launch(void* const* d_in, const int* in_sizes, int n_in,
                              void* d_out, int out_size, void* d_ws, size_t ws_size,
                              hipStream_t stream) {
    const float* X     = (const float*)d_in[0];
    const float* Xfea  = (const float*)d_in[1];
    const float* W1    = (const float*)d_in[2];
    const float* b1    = (const float*)d_in[3];
    const float* gamma = (const float*)d_in[4];
    const float* beta  = (const float*)d_in[5];
    const float* W2    = (const float*)d_in[6];
    const float* b2    = (const float*)d_in[7];
    float* out = (float*)d_out;

    float* ws   = (float*)d_ws;
    float* h    = ws;                          // N*FEA
    float* sums = h + NPTS * FEA;              // 64
    float* wall = sums + 64;                   // NITER * N * NBW
    float4* Xs  = (float4*)(wall + NITER * NPTS * NBW);   // N float4
    float* Xa   = (float*)(Xs + NPTS);         // N*3
    float* Xb   = Xa + NPTS * 3;               // N*3

    // ---- bandwidth weights for all iterations (X_fea is iteration-invariant) ----
    for (int i = 0; i < NITER; i++) {
        ms_zero_kernel<<<1, 64, 0, stream>>>(sums, 64);
        ms_mlp1_kernel<<<NPTS / 256, 256, 0, stream>>>(
            Xfea, W1 + i * FEA * FEA, b1 + i * FEA, h, sums);
        ms_mlp2_kernel<<<NPTS / 256, 256, 0, stream>>>(
            h, sums, gamma + i * FEA, beta + i * FEA,
            W2 + i * FEA * NBW, b2 + i * NBW, wall + i * NPTS * NBW);
    }

    // ---- meanshift chain: d_in -> Xa -> Xb -> Xa -> d_out ----
    const float* cur = X;
    for (int i = 0; i < NITER; i++) {
        float* nxt = (i == NITER - 1) ? out : ((i & 1) ? Xb : Xa);
        ms_pack_kernel<<<NPTS / 256, 256, 0, stream>>>(cur, Xs);
        ms_zero_kernel<<<(NPTS * 3 + 255) / 256, 256, 0, stream>>>(nxt, NPTS * 3);
        // 384 i-tiles * 4 bands = 1536 waves = 192 blocks of 256 threads
        ms_step_kernel<<<(NPTS / 16) * NBW * 32 / 256, 256, 0, stream>>>(
            Xs, wall + i * NPTS * NBW, nxt);
        cur = nxt;
    }
}